// quantile_activation_1d_65240553226664
// MI455X (gfx1250) — compile-verified
//
#include <hip/hip_runtime.h>
#include <stdint.h>

#define CVALF   100.0f
#define BN_EPSF 1e-5f

#define QA_THREADS 512
#define SORT_N     1024          // >= Ns+2, power of two

#define CTILE        128         // channels per block in main kernel
#define ROWS_PER_BLK 64
#define MB_THREADS   256

#if __has_builtin(__builtin_amdgcn_tensor_load_to_lds) && \
    __has_builtin(__builtin_amdgcn_s_wait_tensorcnt)
#define USE_TDM 1
typedef unsigned int tdm_u32x4 __attribute__((ext_vector_type(4)));
typedef int          tdm_i32x8 __attribute__((ext_vector_type(8)));
typedef int          tdm_i32x4 __attribute__((ext_vector_type(4)));
#else
#define USE_TDM 0
#endif

// ---------------------------------------------------------------------------
// Kernel B (defined first so it leads the asm listing): streaming
// quantile-activation. Memory-bound (~256 MB total => ~11 us at 23.3 TB/s).
// Tile of 128 channels' quantile rows staged into LDS via the Tensor Data
// Mover; per-element branchless binary search (7 LDS probes, conflict-free:
// stride 102 dwords => 32 distinct banks across a wave32).
// ---------------------------------------------------------------------------
__global__ __launch_bounds__(MB_THREADS)
void qact_kernel(const float* __restrict__ x,
                 const float* __restrict__ gi_g, const float* __restrict__ gi_b,
                 const float* __restrict__ gi_m, const float* __restrict__ gi_v,
                 const float* __restrict__ go_g, const float* __restrict__ go_b,
                 const float* __restrict__ go_m, const float* __restrict__ go_v,
                 const float* __restrict__ quantiles,
                 const float* __restrict__ quant_list,
                 float* __restrict__ out,
                 int B, int Nc, int K) {
  __shared__ float sQ[CTILE * 102];   // quantile rows, stride K dwords
  __shared__ float sQL[102];
  const int tid = threadIdx.x;
  const int c0  = blockIdx.x * CTILE;
  const int r0  = blockIdx.y * ROWS_PER_BLK;

  if (tid < K) sQL[tid] = quant_list[tid];

#if USE_TDM
  if (tid < 32) {  // wave 0 only: TDM ignores EXEC, gate at wave granularity
    const uint32_t nd      = (uint32_t)(CTILE * K);           // dwords, contiguous
    const uint32_t lds_off = (uint32_t)(uintptr_t)(void*)&sQ[0];
    const uint64_t ga      = (uint64_t)(uintptr_t)(quantiles + (size_t)c0 * K);
    // D# group 0: count=1 | lds_addr | global_addr[56:0] | type=2
    tdm_u32x4 g0 = { 1u,
                     lds_off,
                     (uint32_t)ga,
                     (uint32_t)((ga >> 32) & 0x01FFFFFFull) | 0x80000000u };
    // D# group 1: data_size=4B, 1-D tile: tensor_dim0 = tile_dim0 = nd
    tdm_i32x8 g1 = { (int)(2u << 16),                              // data_size=2 (4B)
                     (int)((nd & 0xFFFFu) << 16),                  // tensor_dim0 lo
                     (int)(((nd >> 16) & 0xFFFFu) | (1u << 16)),   // dim0 hi | tensor_dim1=1
                     (int)((nd & 0xFFFFu) << 16),                  // tile_dim0
                     0,                                            // tile_dim1/2 unused
                     (int)nd,                                      // tensor_dim0_stride lo
                     0, 0 };
    tdm_i32x4 g2 = { 0, 0, 0, 0 };
    tdm_i32x4 g3 = { 0, 0, 0, 0 };
    tdm_i32x8 g4 = { 0, 0, 0, 0, 0, 0, 0, 0 };  // extra group (6-arg toolchain)
    __builtin_amdgcn_tensor_load_to_lds(g0, g1, g2, g3, g4, 0);
    __builtin_amdgcn_s_wait_tensorcnt(0);
  }
#else
  for (int i = tid; i < CTILE * K; i += MB_THREADS)
    sQ[i] = quantiles[(size_t)c0 * K + i];
#endif
  __syncthreads();

  const int c_local = tid & (CTILE - 1);
  const int c = c0 + c_local;
  const float si = gi_g[c] * rsqrtf(gi_v[c] + BN_EPSF);
  const float bi = gi_b[c] - gi_m[c] * si;
  const float so = go_g[c] * rsqrtf(go_v[c] + BN_EPSF);
  const float bo = go_b[c] - go_m[c] * so;
  const float* qrow = &sQ[c_local * K];

  const int rstride = MB_THREADS / CTILE;  // 2
  for (int r = r0 + (tid >> 7); r < r0 + ROWS_PER_BLK; r += rstride) {
    if (r >= B) break;
    const size_t off = (size_t)r * Nc + c;
    if (r + rstride < B)
      __builtin_prefetch(&x[off + (size_t)rstride * Nc], 0, 0); // global_prefetch_b8
    const float z = fmaf(x[off], si, bi);
    // searchsorted(qrow, z, 'right'), branchless over K<=127
    int cnt = 0;
    #pragma unroll
    for (int step = 64; step > 0; step >>= 1) {
      const int t2 = cnt + step;
      if (t2 <= K && qrow[t2 - 1] <= z) cnt = t2;
    }
    const int idx = min(cnt, K - 1);
    out[off] = fmaf(sQL[idx], so, bo);
  }
}

// ---------------------------------------------------------------------------
// Kernel A: per-channel weighted quantiles.
// One block per channel. Sort the 1002 padded values (bitonic, LDS, padded to
// 1024 with +inf). Cumulative weights are analytic since weight depends only
// on sign and negatives sort first: cw[i<sn]=(i+1)*wneg, else base+(i+1-sn)*wpos.
// ---------------------------------------------------------------------------
__global__ __launch_bounds__(QA_THREADS)
void wq_kernel(const float* __restrict__ context,
               const float* __restrict__ quant_list,
               float* __restrict__ quantiles,
               int Ns, int K) {
  __shared__ float sv[SORT_N];
  __shared__ float cw[SORT_N];
  __shared__ int s_neg;
  const int c   = blockIdx.x;
  const int tid = threadIdx.x;
  if (tid == 0) s_neg = 0;
  __syncthreads();

  int negc = 0;
  const float* row = context + (size_t)c * Ns;
  for (int i = tid; i < SORT_N; i += QA_THREADS) {
    float v;
    if (i < Ns)          { v = row[i]; negc += (v < 0.0f) ? 1 : 0; }
    else if (i == Ns)      v = -CVALF;           // pad_lo
    else if (i == Ns + 1)  v =  CVALF;           // pad_hi
    else                   v = __builtin_inff(); // sort filler
    sv[i] = v;
  }
  if (negc) atomicAdd(&s_neg, negc);
  __syncthreads();

  const int total_n = Ns + 2;
  const int sn = s_neg + 1;        // negatives incl. -CVAL pad
  const int sp = total_n - sn;     // non-negatives incl. +CVAL pad

  // bitonic sort ascending over SORT_N elements
  for (int k = 2; k <= SORT_N; k <<= 1) {
    for (int j = k >> 1; j > 0; j >>= 1) {
      for (int i = tid; i < SORT_N; i += QA_THREADS) {
        const int ixj = i ^ j;
        if (ixj > i) {
          const float a = sv[i];
          const float b = sv[ixj];
          const bool up = ((i & k) == 0);
          if ((a > b) == up) { sv[i] = b; sv[ixj] = a; }
        }
      }
      __syncthreads();
    }
  }

  const float wneg   = (float)Ns / (float)sn;
  const float wpos   = (float)Ns / (float)sp;
  const float negTot = (float)sn * wneg;
  for (int i = tid; i < total_n; i += QA_THREADS)
    cw[i] = (i < sn) ? (float)(i + 1) * wneg
                     : negTot + (float)(i + 1 - sn) * wpos;
  __syncthreads();

  const float total = negTot + (float)sp * wpos;   // == cw[total_n-1]
  if (tid < K) {
    const float tw = quant_list[tid] * total;
    // searchsorted(cw, tw, 'right') = count of cw[j] <= tw  (branchless)
    int cnt = 0;
    #pragma unroll
    for (int step = 512; step > 0; step >>= 1) {
      const int t2 = cnt + step;
      if (t2 <= total_n && cw[t2 - 1] <= tw) cnt = t2;
    }
    const int below = min(max(cnt - 1, 0), Ns - 1);
    const int above = min(cnt, Ns - 1);
    const float vb = sv[below], va = sv[above];
    const float wb = cw[below], wa = cw[above];
    const float frac = (tw - wb) / (wa - wb + 1e-6f);
    quantiles[(size_t)c * K + tid] = fmaf(va - vb, frac, vb);
  }
}

// ---------------------------------------------------------------------------
extern "C" void kernel_launch(void* const* d_in, const int* in_sizes, int n_in,
                              void* d_out, int out_size, void* d_ws, size_t ws_size,
                              hipStream_t stream) {
  (void)n_in; (void)out_size; (void)ws_size;
  const float* x          = (const float*)d_in[0];
  const float* gi_g       = (const float*)d_in[1];
  const float* gi_b       = (const float*)d_in[2];
  const float* gi_m       = (const float*)d_in[3];
  const float* gi_v       = (const float*)d_in[4];
  const float* go_g       = (const float*)d_in[5];
  const float* go_b       = (const float*)d_in[6];
  const float* go_m       = (const float*)d_in[7];
  const float* go_v       = (const float*)d_in[8];
  const float* context    = (const float*)d_in[9];
  const float* quant_list = (const float*)d_in[10];
  float* outp = (float*)d_out;

  const int Nc = in_sizes[1];
  const int B  = in_sizes[0] / Nc;
  const int Ns = in_sizes[9] / Nc;
  const int K  = in_sizes[10];

  float* quantiles = (float*)d_ws;   // Nc*K floats of scratch

  wq_kernel<<<Nc, QA_THREADS, 0, stream>>>(context, quant_list, quantiles, Ns, K);

  dim3 grid(Nc / CTILE, (B + ROWS_PER_BLK - 1) / ROWS_PER_BLK);
  qact_kernel<<<grid, MB_THREADS, 0, stream>>>(x, gi_g, gi_b, gi_m, gi_v,
                                               go_g, go_b, go_m, go_v,
                                               quantiles, quant_list, outp,
                                               B, Nc, K);
}